// QueryAndGroup_4990751998370
// MI455X (gfx1250) — compile-verified
//
#include <hip/hip_runtime.h>

// Problem constants (match reference)
constexpr int kB      = 4;
constexpr int kN      = 16384;
constexpr int kNP     = 2048;   // NPOINT
constexpr int kC      = 64;     // feature channels
constexpr int kNS     = 32;     // NSAMPLE
constexpr int kNCH    = kC + 3; // 67 output channels
constexpr float kR2   = 0.2f * 0.2f;

typedef __attribute__((ext_vector_type(2))) float v2f;
typedef __attribute__((ext_vector_type(8))) float v8f;

// ---------------------------------------------------------------------------
// Kernel 1: pack xyz -> (x, y, z, |p|^2) float4, once per batch point.
// ---------------------------------------------------------------------------
__global__ void pack_kernel(const float* __restrict__ xyz,
                            float4* __restrict__ packed) {
  int gid = blockIdx.x * blockDim.x + threadIdx.x;   // < kB * kN
  float x = xyz[3 * gid + 0];
  float y = xyz[3 * gid + 1];
  float z = xyz[3 * gid + 2];
  packed[gid] = make_float4(x, y, z, x * x + y * y + z * z);
}

// ---------------------------------------------------------------------------
// Kernel 2: ball query via V_WMMA_F32_16X16X4_F32.
// One wave per block handles 16 query points; loops over N in tiles of 16.
//   D[m][n] = (-2 q_m) . p_n + |p_n|^2 + (|q_m|^2 - r^2) = |p_n - q_m|^2 - r^2
// In-ball mask = (D < 0) via ballot. Compaction is 16-lane parallel:
// lane r owns matrix row r, appends ascending set-bit indices into a
// bank-conflict-free transposed LDS list, capped at 32; early exit once all
// 16 rows are full (single ballot test).
// ---------------------------------------------------------------------------
__global__ void __launch_bounds__(32)
ballquery_kernel(const float4* __restrict__ packed,
                 const float* __restrict__ new_xyz,
                 int* __restrict__ idx_out) {
  const int lane = threadIdx.x;
  const int b  = blockIdx.x / (kNP / 16);
  const int j0 = (blockIdx.x % (kNP / 16)) * 16;

  // Load this wave's 16 query points (lanes 16-31 mirror lanes 0-15).
  const int qi = j0 + (lane & 15);
  const float qx = new_xyz[((size_t)b * kNP + qi) * 3 + 0];
  const float qy = new_xyz[((size_t)b * kNP + qi) * 3 + 1];
  const float qz = new_xyz[((size_t)b * kNP + qi) * 3 + 2];

  // A matrix 16x4 (2 VGPRs): row m = (-2qx, -2qy, -2qz, 1)
  //   VGPR0: lanes 0-15 K=0, lanes 16-31 K=2 ; VGPR1: K=1 / K=3
  v2f Am;
  Am.x = (lane < 16) ? (-2.0f * qx) : (-2.0f * qz);
  Am.y = (lane < 16) ? (-2.0f * qy) : 1.0f;

  // C matrix 16x16 (8 VGPRs): C[m][n] = |q_m|^2 - r^2 (constant per row).
  //   VGPR v: lanes 0-15 -> row v, lanes 16-31 -> row v+8.
  const float qn = qx * qx + qy * qy + qz * qz - kR2;
  v8f Cm;
#pragma unroll
  for (int v = 0; v < 8; ++v) {
    float lo = __shfl(qn, v, 32);
    float hi = __shfl(qn, v + 8, 32);
    Cm[v] = (lane < 16) ? lo : hi;
  }

  // Transposed index list: sIdx[slot*16 + row] -> 16 lanes append to 16
  // consecutive DWORD banks simultaneously (no LDS bank conflicts).
  __shared__ int sIdx[kNS * 16];

  // Per-lane append count: lane r (r<16) counts row r; upper lanes start
  // "full" so the all-done ballot is a single compare.
  int cnt = (lane < 16) ? 0 : kNS;

  const float4* pb = packed + (size_t)b * kN;
  bool doneAll = false;

  for (int base = 0; base < kN && !doneAll; base += 16) {
    // B matrix 4x16 (2 VGPRs): column n = (px, py, pz, |p|^2)
    float4 p = pb[base + (lane & 15)];
    int pf = base + 64 + (lane & 15);
    if (pf < kN) __builtin_prefetch((const void*)(pb + pf), 0, 1);

    v2f Bm;
    Bm.x = (lane < 16) ? p.x : p.z;
    Bm.y = (lane < 16) ? p.y : p.w;

    // D = A*B + C  ->  v_wmma_f32_16x16x4_f32
    v8f D = __builtin_amdgcn_wmma_f32_16x16x4_f32(
        false, Am, false, Bm, (short)0, Cm, false, false);

    // Ballot each D VGPR: bits[0:15] = row v, bits[16:31] = row v+8.
    unsigned msk[8];
#pragma unroll
    for (int v = 0; v < 8; ++v)
      msk[v] = __builtin_amdgcn_ballot_w32(D[v] < 0.0f);

    // Redistribute: lane v (v<8) holds msk[v]; then lane r (r<16) grabs its
    // row mask via one shuffle + hi/lo select.
    unsigned tmp = 0;
#pragma unroll
    for (int v = 0; v < 8; ++v)
      tmp = (lane == v) ? msk[v] : tmp;
    unsigned rowm = (unsigned)__shfl((int)tmp, lane & 7, 32);
    unsigned bits = (lane < 16) ? ((lane < 8) ? (rowm & 0xFFFFu)
                                              : (rowm >> 16))
                                : 0u;

    // 16-lane parallel compaction, ascending index order per row.
    while (bits && cnt < kNS) {
      int pos = __builtin_ctz(bits);
      sIdx[cnt * 16 + lane] = base + pos;
      cnt++;
      bits &= bits - 1;
    }

    doneAll = (__builtin_amdgcn_ballot_w32(cnt >= kNS) == 0xFFFFFFFFu);
  }

  __syncthreads();

  // Write idx[b][j0+r][0..31]; pad with first hit, or 0 if no hit (matches
  // reference: order[...,0] == 0 when mask is empty).
#pragma unroll 1
  for (int r = 0; r < 16; ++r) {
    int n = __shfl(cnt, r, 32);
    int first = (n > 0) ? sIdx[0 * 16 + r] : 0;
    int val = (lane < n) ? sIdx[lane * 16 + r] : first;
    idx_out[(size_t)(b * kNP + j0 + r) * kNS + lane] = val;
  }
}

// ---------------------------------------------------------------------------
// Kernel 3: gather + center. One thread per output element of
// (B, 67, NPOINT, 32). ch<3 -> xyz gather minus query; else feature gather.
// ---------------------------------------------------------------------------
__global__ void gather_kernel(const float* __restrict__ xyz,
                              const float* __restrict__ new_xyz,
                              const float* __restrict__ feat,
                              const int* __restrict__ idx,
                              float* __restrict__ out, size_t total) {
  size_t tid = (size_t)blockIdx.x * blockDim.x + threadIdx.x;
  if (tid >= total) return;
  int k  = (int)(tid & (kNS - 1));
  int j  = (int)((tid >> 5) & (kNP - 1));
  int bc = (int)(tid >> 16);                 // kNS*kNP == 65536 == 1<<16
  int ch = bc % kNCH;
  int b  = bc / kNCH;
  int i  = idx[((size_t)b * kNP + j) * kNS + k];
  float val;
  if (ch < 3)
    val = xyz[((size_t)b * kN + i) * 3 + ch] -
          new_xyz[((size_t)b * kNP + j) * 3 + ch];
  else
    val = feat[((size_t)b * kC + (ch - 3)) * kN + i];
  out[tid] = val;
}

// ---------------------------------------------------------------------------
extern "C" void kernel_launch(void* const* d_in, const int* in_sizes, int n_in,
                              void* d_out, int out_size, void* d_ws, size_t ws_size,
                              hipStream_t stream) {
  const float* xyz     = (const float*)d_in[0];  // (B, N, 3)
  const float* new_xyz = (const float*)d_in[1];  // (B, NPOINT, 3)
  const float* feat    = (const float*)d_in[2];  // (B, C, N)

  // Workspace layout: [packed float4: B*N*16 B = 1 MB][idx: B*NPOINT*32*4 B = 1 MB]
  float4* packed = (float4*)d_ws;
  int* idxbuf = (int*)((char*)d_ws + (size_t)kB * kN * sizeof(float4));

  pack_kernel<<<(kB * kN) / 256, 256, 0, stream>>>(xyz, packed);
  ballquery_kernel<<<kB * (kNP / 16), 32, 0, stream>>>(packed, new_xyz, idxbuf);

  size_t total = (size_t)kB * kNCH * kNP * kNS;   // 17,563,648
  gather_kernel<<<(unsigned)((total + 255) / 256), 256, 0, stream>>>(
      xyz, new_xyz, feat, idxbuf, (float*)d_out, total);
}